// DeepFMModel_33672543600867
// MI455X (gfx1250) — compile-verified
//
#include <hip/hip_runtime.h>

// DeepFM fused forward for MI455X (gfx1250, wave32).
// One wave handles 16 batch rows end-to-end: embedding gather -> LDS,
// FM + linear on lanes 0-15, then the MLP chain via V_WMMA_F32_16X16X4_F32
// (fp32 WMMA, exact vs the fp32 reference). All weights stream from L2.

#define NFIELD 8
#define EDIM   16
#define DNN_IN 128
#define LDW    132   // LDS row stride (floats): 132 % 64 == 4 -> conflict-free frag reads

typedef __attribute__((ext_vector_type(2))) float v2f;
typedef __attribute__((ext_vector_type(8))) float v8f;

__device__ __constant__ int kOffsets[NFIELD] = {
    0, 50000, 55000, 57000, 58000, 59000, 59500, 59800
};

static __device__ __forceinline__ v8f wmma4(v2f a, v2f b, v8f c) {
    // D = A(16x4 f32) * B(4x16 f32) + C(16x16 f32)
    return __builtin_amdgcn_wmma_f32_16x16x4_f32(
        /*neg_a=*/false, a, /*neg_b=*/false, b,
        /*c_mod=*/(short)0, c, /*reuse_a=*/false, /*reuse_b=*/false);
}

// A-fragment from a row-major LDS tile (width LDW), K-step base k:
// lanes 0-15 hold row (lane&15), K = k+{0,1}; lanes 16-31 hold K = k+{2,3}.
static __device__ __forceinline__ v2f load_a(const float* buf, int lm, int half, int k) {
    const float* p = buf + lm * LDW + k + 2 * half;
    v2f a; a.x = p[0]; a.y = p[1];
    return a;
}

// B-fragment from a row-major global weight W[K][N] (leading dim ldn):
// lanes 0-15: rows k+{0,1}; lanes 16-31: rows k+{2,3}; col = ncol + (lane&15).
static __device__ __forceinline__ v2f load_b(const float* __restrict__ W, int ldn,
                                             int k, int half, int ncol, int lm) {
    const float* p = W + (k + 2 * half) * ldn + ncol + lm;
    v2f b; b.x = p[0]; b.y = p[ldn];
    return b;
}

__global__ __launch_bounds__(64)
void deepfm_fused_kernel(const int*   __restrict__ x,       // [B,8] int32
                         const float* __restrict__ emb,     // [60000,16]
                         const float* __restrict__ wlin,    // [60000]
                         const float* __restrict__ blin,    // [1]
                         const float* __restrict__ w1,      // [128,256]
                         const float* __restrict__ b1,      // [256]
                         const float* __restrict__ w2,      // [256,128]
                         const float* __restrict__ b2,      // [128]
                         const float* __restrict__ w3,      // [128,64]
                         const float* __restrict__ b3,      // [64]
                         const float* __restrict__ wlast,   // [64]
                         float*       __restrict__ out)     // [B]
{
    __shared__ __align__(16) float smem[2][2][16 * LDW];   // [wave][buf] 33792 B

    const int tid  = threadIdx.x;
    const int wave = tid >> 5;
    const int lane = tid & 31;
    const int half = lane >> 4;   // 0: K pair {0,1} / rows 0-7 ; 1: {2,3} / rows 8-15
    const int lm   = lane & 15;   // A row / B,C column within tile
    const int rowBase = blockIdx.x * 32 + wave * 16;

    float* buf0 = smem[wave][0];
    float* buf1 = smem[wave][1];

    // ---- Embedding gather: 16 rows x 8 fields x 16 f32 -> buf0[row][f*16+d]
    {
        const int r    = lane >> 1;           // 0..15
        const int f0   = (lane & 1) * 4;      // fields f0..f0+3
        const int rowG = rowBase + r;
        #pragma unroll
        for (int ff = 0; ff < 4; ++ff) {
            const int f   = f0 + ff;
            const int idx = x[rowG * NFIELD + f] + kOffsets[f];
            const float4* src = (const float4*)(emb + (long)idx * EDIM);
            float4* dst = (float4*)(buf0 + r * LDW + f * EDIM);
            dst[0] = src[0]; dst[1] = src[1]; dst[2] = src[2]; dst[3] = src[3];
        }
    }
    __syncthreads();

    // ---- FM second-order + linear term: lanes 0-15, one row each
    float yfm = 0.f;
    if (lane < 16) {
        const int rowG = rowBase + lane;
        const float* h = buf0 + lane * LDW;
        float fm = 0.f;
        for (int d = 0; d < EDIM; ++d) {
            float s = 0.f, ss = 0.f;
            #pragma unroll
            for (int f = 0; f < NFIELD; ++f) {
                const float v = h[f * EDIM + d];
                s += v; ss += v * v;
            }
            fm += s * s - ss;
        }
        float lin = blin[0];
        #pragma unroll
        for (int f = 0; f < NFIELD; ++f)
            lin += wlin[x[rowG * NFIELD + f] + kOffsets[f]];
        yfm = lin + 0.5f * fm;
    }
    // (no buf writes since last sync; safe to proceed)

    const v8f vzero = {0.f, 0.f, 0.f, 0.f, 0.f, 0.f, 0.f, 0.f};
    v8f acc2[8];
    #pragma unroll
    for (int t = 0; t < 8; ++t) acc2[t] = vzero;

    // ---- Layer1 (K=128 -> N=256) interleaved with Layer2 (K=256 -> N=128)
    // in two K=128 chunks so h1 never needs more than a 16x128 LDS tile.
    for (int pass = 0; pass < 2; ++pass) {
        // Layer1 tiles nt = pass*8 .. pass*8+7 -> relu(h1 chunk) into buf1
        for (int t = 0; t < 8; ++t) {
            const int nt = pass * 8 + t;
            v8f acc = vzero;
            for (int k = 0; k < DNN_IN; k += 4) {
                const v2f a = load_a(buf0, lm, half, k);
                const v2f b = load_b(w1, 256, k, half, nt * 16, lm);
                acc = wmma4(a, b, acc);
            }
            const float bias = b1[nt * 16 + lm];
            #pragma unroll
            for (int v = 0; v < 8; ++v) {
                float val = acc[v] + bias;
                val = val > 0.f ? val : 0.f;
                buf1[(v + 8 * half) * LDW + t * 16 + lm] = val;
            }
        }
        __syncthreads();

        // Layer2 partial accumulation over this K-chunk (A-fragment reused x8)
        for (int k = 0; k < 128; k += 4) {
            const v2f a = load_a(buf1, lm, half, k);
            #pragma unroll
            for (int t2 = 0; t2 < 8; ++t2) {
                const v2f b = load_b(w2, 128, pass * 128 + k, half, t2 * 16, lm);
                acc2[t2] = wmma4(a, b, acc2[t2]);
            }
        }
        __syncthreads();
    }

    // ---- h2 = relu(acc2 + b2) -> buf0 (16 x 128); buf0 (h0) is dead now
    #pragma unroll
    for (int t2 = 0; t2 < 8; ++t2) {
        const float bias = b2[t2 * 16 + lm];
        #pragma unroll
        for (int v = 0; v < 8; ++v) {
            float val = acc2[t2][v] + bias;
            val = val > 0.f ? val : 0.f;
            buf0[(v + 8 * half) * LDW + t2 * 16 + lm] = val;
        }
    }
    __syncthreads();

    // ---- Layer3 (K=128 -> N=64): relu(h3) -> buf1 (16 x 64)
    for (int t3 = 0; t3 < 4; ++t3) {
        v8f acc = vzero;
        for (int k = 0; k < 128; k += 4) {
            const v2f a = load_a(buf0, lm, half, k);
            const v2f b = load_b(w3, 64, k, half, t3 * 16, lm);
            acc = wmma4(a, b, acc);
        }
        const float bias = b3[t3 * 16 + lm];
        #pragma unroll
        for (int v = 0; v < 8; ++v) {
            float val = acc[v] + bias;
            val = val > 0.f ? val : 0.f;
            buf1[(v + 8 * half) * LDW + t3 * 16 + lm] = val;
        }
    }
    __syncthreads();

    // ---- Final 64->1 dot product + combine with FM/linear
    if (lane < 16) {
        const float* h = buf1 + lane * LDW;
        float y = 0.f;
        #pragma unroll
        for (int k = 0; k < 64; ++k) y += h[k] * wlast[k];
        out[rowBase + lane] = yfm + y;
    }
}

extern "C" void kernel_launch(void* const* d_in, const int* in_sizes, int n_in,
                              void* d_out, int out_size, void* d_ws, size_t ws_size,
                              hipStream_t stream) {
    (void)in_sizes; (void)n_in; (void)d_ws; (void)ws_size;
    const int batch = out_size;                 // 2048
    dim3 grid(batch / 32), block(64);           // 2 waves/block, 16 rows/wave
    deepfm_fused_kernel<<<grid, block, 0, stream>>>(
        (const int*)  d_in[0],   // x
        (const float*)d_in[1],   // emb_table
        (const float*)d_in[2],   // w_lin
        (const float*)d_in[3],   // b_lin
        (const float*)d_in[4],   // w1
        (const float*)d_in[5],   // b1
        (const float*)d_in[6],   // w2
        (const float*)d_in[7],   // b2
        (const float*)d_in[8],   // w3
        (const float*)d_in[9],   // b3
        (const float*)d_in[10],  // w_last
        (float*)d_out);
}